// Siglip2VisionEmbeddings_86792699117657
// MI455X (gfx1250) — compile-verified
//
#include <hip/hip_runtime.h>
#include <hip/hip_bf16.h>

typedef __attribute__((ext_vector_type(2))) float v2f;
typedef __attribute__((ext_vector_type(8))) float v8f;

#define B_    32
#define P_    1024
#define D_    768
#define GRID_ 16

#define BM 128
#define BN 64
#define BK 32
#define LDA 36   // A tile row stride (floats): 16B-aligned rows + bank stagger
#define LDB 34   // Bt tile row stride (floats): n-major, k-contiguous; even => 8B-aligned pairs

__device__ __forceinline__ int imin(int a, int b) { return a < b ? a : b; }
__device__ __forceinline__ int imax(int a, int b) { return a > b ? a : b; }

// gfx1250 async global->LDS copy (no VGPR staging, tracked by ASYNCcnt).
// LDS byte address = low 32 bits of the generic pointer (ISA §10.2).
__device__ __forceinline__ void async_copy_b128(uint32_t lds_byte, const float* g) {
  asm volatile("global_load_async_to_lds_b128 %0, %1, off"
               :: "v"(lds_byte), "v"(g) : "memory");
}

// ---------------------------------------------------------------------------
// Kernel 1: per-sample (h, w) = max(position_ids, axis=1) + 1
// ---------------------------------------------------------------------------
__global__ __launch_bounds__(256) void hw_kernel(const int* __restrict__ pos_ids,
                                                 int* __restrict__ hw) {
  __shared__ int sy[256], sx[256];
  const int i = blockIdx.x;
  const int t = threadIdx.x;
  int my = 0, mx = 0;
  const int* base = pos_ids + i * P_ * 2;
  for (int p = t; p < P_; p += 256) {
    my = imax(my, base[p * 2 + 0]);
    mx = imax(mx, base[p * 2 + 1]);
  }
  sy[t] = my;
  sx[t] = mx;
  __syncthreads();
  for (int s = 128; s > 0; s >>= 1) {
    if (t < s) {
      sy[t] = imax(sy[t], sy[t + s]);
      sx[t] = imax(sx[t], sx[t + s]);
    }
    __syncthreads();
  }
  if (t == 0) {
    hw[i * 2 + 0] = sy[0] + 1;
    hw[i * 2 + 1] = sx[0] + 1;
  }
}

// ---------------------------------------------------------------------------
// Kernel 2: out = X @ W + b + bilinear_pos_sample (fused epilogue)
// 256 threads = 8 wave32; tile 128x64, K chunks of 32, double-buffered LDS.
// A tile: async global->LDS (4x b128/thread, ASYNCcnt pipeline, no staging VGPRs).
// B tile: register-staged transpose (4x float2/thread) into n-major LDS so each
//         WMMA fragment is one 8B-aligned pair -> ds_load_2addr_b64, no movs.
// ---------------------------------------------------------------------------
__global__ __launch_bounds__(256) void gemm_pos_kernel(
    const float* __restrict__ X, const float* __restrict__ W,
    const float* __restrict__ bias, const float* __restrict__ grid,
    const int* __restrict__ pos_ids, const int* __restrict__ hw,
    float* __restrict__ out) {
  __shared__ float Asm[2][BM * LDA];
  __shared__ float Bsm[2][BN * LDB];

  const int t = threadIdx.x;
  const int lane = t & 31;
  const int wave = t >> 5;
  const int mBlock = blockIdx.x * BM;
  const int nBlock = blockIdx.y * BN;

  const float* Ag = X + (size_t)mBlock * D_;  // row-major, k contiguous
  const float* Bg = W + nBlock;               // k-major rows of 768

  v8f acc[4] = {};    // 16x64 strip per wave
  float2 bReg[4];     // B staging: 32x64 tile = 1024 float2 / 256 thr = 4

  auto issueAsyncA = [&](int kBase, int buf) {  // 4 async b128 per thread
    const uint32_t aB = (uint32_t)(uintptr_t)&Asm[buf][0];
#pragma unroll
    for (int j = 0; j < 4; ++j) {
      int idx = t + 256 * j;  // 16B-unit index, 8 per A row
      int row = idx >> 3, c4 = idx & 7;
      async_copy_b128(aB + (uint32_t)(row * LDA + c4 * 4) * 4,
                      Ag + (size_t)row * D_ + kBase + c4 * 4);
    }
  };
  auto loadB = [&](int kBase) {  // 32x64 tile as 1024 float2; 4 per thread
#pragma unroll
    for (int j = 0; j < 4; ++j) {
      int idx = t + 256 * j;
      int k = idx >> 5, n2 = idx & 31;  // 32 float2 per k-row
      bReg[j] = *(const float2*)(Bg + (size_t)(kBase + k) * D_ + n2 * 2);
    }
  };
  auto storeB = [&](int buf) {  // transpose into n-major LDS
#pragma unroll
    for (int j = 0; j < 4; ++j) {
      int idx = t + 256 * j;
      int k = idx >> 5, n = (idx & 31) * 2;
      Bsm[buf][(n + 0) * LDB + k] = bReg[j].x;
      Bsm[buf][(n + 1) * LDB + k] = bReg[j].y;
    }
  };

  // prologue: A chunks 0,1 in flight; B chunk 0 in LDS, chunk 1 in regs
  issueAsyncA(0, 0);
  issueAsyncA(BK, 1);
  loadB(0);
  storeB(0);
  loadB(BK);
  asm volatile("s_wait_asynccnt 4" ::: "memory");  // A chunk 0 landed (in-order)
  __syncthreads();

  // WMMA 16x16x4 f32 fragments:
  //   A: lanes 0-15 -> M=lane, K={0,1}; lanes 16-31 -> M=lane-16, K={2,3}; B mirrored
  const int mWave = wave * 16;
  const int aRow = mWave + (lane & 15);
  const int kSel = (lane >> 4) << 1;  // 0 or 2
  const int nFrag = lane & 15;

  const int nChunks = D_ / BK;  // 24
  for (int c = 0; c < nChunks; ++c) {
    const int cur = c & 1;
    const float* As = &Asm[cur][0];
    const float* Bs = &Bsm[cur][0];
#pragma unroll
    for (int s = 0; s < 8; ++s) {
      const int k0 = 4 * s + kSel;
      v2f a = *(const v2f*)&As[aRow * LDA + k0];
#pragma unroll
      for (int q = 0; q < 4; ++q) {
        const int n = q * 16 + nFrag;
        v2f b = *(const v2f*)&Bs[n * LDB + k0];
        acc[q] = __builtin_amdgcn_wmma_f32_16x16x4_f32(
            false, a, false, b, (short)0, acc[q], false, false);
      }
    }
    __syncthreads();  // all waves done reading A[cur], B[cur]
    if (c + 2 < nChunks) issueAsyncA((c + 2) * BK, cur);
    if (c + 1 < nChunks) storeB(cur ^ 1);     // bReg holds chunk c+1
    if (c + 2 < nChunks) {
      loadB((c + 2) * BK);
      asm volatile("s_wait_asynccnt 4" ::: "memory");  // A chunk c+1 landed
    } else {
      asm volatile("s_wait_asynccnt 0" ::: "memory");
    }
    __syncthreads();
  }

  // ------------------- fused epilogue: bias + bilinear position embedding ---
  // BM=128 divides P=1024, so the whole block lives in one sample.
  const int sample = mBlock / P_;
  const int h = hw[sample * 2 + 0];
  const int w = hw[sample * 2 + 1];
  // all SIZES >= (16,16): pure upscale => antialias == plain half-pixel bilinear
  const float sy = (float)GRID_ / (float)h;
  const float sx = (float)GRID_ / (float)w;
  const int nLane = nBlock + nFrag;
  const int rowHalf = (lane >> 4) << 3;  // C/D layout: lanes 16-31 hold rows +8

  float biasv[4];
#pragma unroll
  for (int q = 0; q < 4; ++q) biasv[q] = bias[nLane + q * 16];

#pragma unroll
  for (int r = 0; r < 8; ++r) {
    const int rowGlobal = mBlock + mWave + rowHalf + r;
    const int p = rowGlobal & (P_ - 1);
    const int y = pos_ids[(sample * P_ + p) * 2 + 0];
    const int x = pos_ids[(sample * P_ + p) * 2 + 1];
    const float fy = ((float)y + 0.5f) * sy - 0.5f;
    const float fx = ((float)x + 0.5f) * sx - 0.5f;
    const float flY = floorf(fy), flX = floorf(fx);
    const float wy = fy - flY, wx = fx - flX;
    const int y0 = (int)flY, x0 = (int)flX;
    const int y0c = imin(imax(y0, 0), GRID_ - 1);
    const int y1c = imin(imax(y0 + 1, 0), GRID_ - 1);
    const int x0c = imin(imax(x0, 0), GRID_ - 1);
    const int x1c = imin(imax(x0 + 1, 0), GRID_ - 1);
    const float* g00 = grid + (size_t)(y0c * GRID_ + x0c) * D_;
    const float* g01 = grid + (size_t)(y0c * GRID_ + x1c) * D_;
    const float* g10 = grid + (size_t)(y1c * GRID_ + x0c) * D_;
    const float* g11 = grid + (size_t)(y1c * GRID_ + x1c) * D_;
    const float w00 = (1.f - wy) * (1.f - wx), w01 = (1.f - wy) * wx;
    const float w10 = wy * (1.f - wx), w11 = wy * wx;
    float* orow = out + (size_t)rowGlobal * D_;
#pragma unroll
    for (int q = 0; q < 4; ++q) {
      const int n = nLane + q * 16;
      const float posv = w00 * g00[n] + w01 * g01[n] + w10 * g10[n] + w11 * g11[n];
      orow[n] = acc[q][r] + biasv[q] + posv;
    }
  }
}

// ---------------------------------------------------------------------------
extern "C" void kernel_launch(void* const* d_in, const int* in_sizes, int n_in,
                              void* d_out, int out_size, void* d_ws, size_t ws_size,
                              hipStream_t stream) {
  const float* pixel_values = (const float*)d_in[0];  // (32,1024,768)
  const int* position_ids = (const int*)d_in[1];      // (32,1024,2)
  const float* patch_W = (const float*)d_in[2];       // (768,768) [in,out]
  const float* patch_b = (const float*)d_in[3];       // (768)
  const float* pos_emb = (const float*)d_in[4];       // (256,768)
  float* out = (float*)d_out;                         // (32,1024,768)
  int* hw = (int*)d_ws;                               // 32 x (h,w)

  hw_kernel<<<B_, 256, 0, stream>>>(position_ids, hw);

  dim3 gridDim((B_ * P_) / BM, D_ / BN);  // 256 x 12
  gemm_pos_kernel<<<gridDim, 256, 0, stream>>>(pixel_values, patch_W, patch_b,
                                               pos_emb, position_ids, hw, out);
}